// RWPEEncoder_59365037965997
// MI455X (gfx1250) — compile-verified
//
#include <hip/hip_runtime.h>

typedef __attribute__((ext_vector_type(2))) float v2f;
typedef __attribute__((ext_vector_type(8))) float v8f;

// ---------------------------------------------------------------------------
// Phase 1: degree computation (self-loop weight 1 folded into the init)
// ---------------------------------------------------------------------------
__global__ void k_init_deg(float* __restrict__ deg, int N) {
    int i = blockIdx.x * blockDim.x + threadIdx.x;
    if (i < N) deg[i] = 1.0f;  // self-loop contributes weight 1 to deg[row]
}

__global__ void k_accum_deg(const int* __restrict__ row, const float* __restrict__ ew,
                            float* __restrict__ deg, int E) {
    int e = blockIdx.x * blockDim.x + threadIdx.x;
    if (e < E) atomicAdd(&deg[row[e]], ew[e]);
}

__global__ void k_finish_deg(float* __restrict__ deg_inv, float* __restrict__ prob0, int N) {
    int i = blockIdx.x * blockDim.x + threadIdx.x;
    if (i < N) {
        float d = deg_inv[i];
        deg_inv[i] = 1.0f / fmaxf(d, 1e-8f);
        prob0[i] = 1.0f;       // initial probability vector
    }
}

// ---------------------------------------------------------------------------
// Phase 2: one power-iteration step
//   next[i]    = prev[i] * deg_inv[i]                    (self loop)
//   next[col] += prev[row] * ew * deg_inv[row]           (edges, L2 atomics)
// ---------------------------------------------------------------------------
__global__ void k_self_step(const float* __restrict__ prev, const float* __restrict__ deg_inv,
                            float* __restrict__ next, int N) {
    int i = blockIdx.x * blockDim.x + threadIdx.x;
    if (i < N) next[i] = prev[i] * deg_inv[i];
}

__global__ void k_scatter(const int* __restrict__ row, const int* __restrict__ col,
                          const float* __restrict__ ew, const float* __restrict__ deg_inv,
                          const float* __restrict__ prev, float* __restrict__ next, int E) {
    int e = blockIdx.x * blockDim.x + threadIdx.x;
    if (e < E) {
        int r = row[e];
        atomicAdd(&next[col[e]], prev[r] * ew[e] * deg_inv[r]);
    }
}

// ---------------------------------------------------------------------------
// Phase 3: MLP with f32 WMMA (V_WMMA_F32_16X16X4_F32), one wave per 16 nodes.
// feat is (16, N): feat[k*N + i] = prob after step k+1 at node i.
// out[i][j] = sum_h W2[j][h] * relu(sum_k W1[h][k]*feat[k][i] + b1[h]) + b2[j]
// ---------------------------------------------------------------------------
__global__ __launch_bounds__(32) void rwpe_mlp_wmma(
    const float* __restrict__ feat,
    const float* __restrict__ W1, const float* __restrict__ b1,
    const float* __restrict__ W2, const float* __restrict__ b2,
    float* __restrict__ out, int N)
{
    __shared__ float hs[16 * 64];          // 16-node hidden tile, 4 KB
    const int lane = threadIdx.x & 31;
    const int half = lane >> 4;            // 0: lanes 0-15, 1: lanes 16-31
    const int l16  = lane & 15;
    const int m0   = blockIdx.x * 16;      // first node of this tile
    const int M    = m0 + l16;             // A-matrix row held by this lane
    const int kb   = half * 2;             // K offset inside a 4-wide chunk

    // ---- Layer 1: h = relu(rwpe @ W1^T + b1), K = 16 (4 chunks of 4) ----
    v2f a1[4];
#pragma unroll
    for (int kk = 0; kk < 4; ++kk) {
        int k0 = kk * 4 + kb;
        a1[kk].x = feat[(size_t)(k0    ) * N + M];
        a1[kk].y = feat[(size_t)(k0 + 1) * N + M];
    }
#pragma unroll
    for (int nt = 0; nt < 4; ++nt) {
        const int n = nt * 16 + l16;       // B/C column held by this lane
        v8f c = {};
#pragma unroll
        for (int kk = 0; kk < 4; ++kk) {
            int k0 = kk * 4 + kb;
            v2f b;
            b.x = W1[n * 16 + k0];         // B[k][n] = W1[n][k]
            b.y = W1[n * 16 + k0 + 1];
            c = __builtin_amdgcn_wmma_f32_16x16x4_f32(false, a1[kk], false, b,
                                                      (short)0, c, false, false);
        }
        const float bias = b1[n];
#pragma unroll
        for (int r = 0; r < 8; ++r) {
            float v = c[r] + bias;
            hs[(r + half * 8) * 64 + n] = v > 0.0f ? v : 0.0f;   // relu
        }
    }
    __syncthreads();   // single-wave WG: dscnt wait + S_NOP barrier

    // ---- Layer 2: out = h @ W2^T + b2, K = 64 (16 chunks of 4) ----
    v2f a2[16];
#pragma unroll
    for (int kk = 0; kk < 16; ++kk) {
        int k0 = kk * 4 + kb;
        a2[kk].x = hs[l16 * 64 + k0];
        a2[kk].y = hs[l16 * 64 + k0 + 1];
    }
#pragma unroll
    for (int nt = 0; nt < 4; ++nt) {
        const int n = nt * 16 + l16;
        v8f c = {};
#pragma unroll
        for (int kk = 0; kk < 16; ++kk) {
            int k0 = kk * 4 + kb;
            v2f b;
            b.x = W2[n * 64 + k0];         // B[k][n] = W2[n][k]
            b.y = W2[n * 64 + k0 + 1];
            c = __builtin_amdgcn_wmma_f32_16x16x4_f32(false, a2[kk], false, b,
                                                      (short)0, c, false, false);
        }
        const float bias = b2[n];
#pragma unroll
        for (int r = 0; r < 8; ++r)
            out[(size_t)(m0 + r + half * 8) * 64 + n] = c[r] + bias;
    }
}

// Scalar tail for node rows beyond the last full 16-tile (unused for N=100000).
__global__ void mlp_tail(const float* __restrict__ feat,
                         const float* __restrict__ W1, const float* __restrict__ b1,
                         const float* __restrict__ W2, const float* __restrict__ b2,
                         float* __restrict__ out, int N, int m0) {
    int t = blockIdx.x * blockDim.x + threadIdx.x;
    int rem = N - m0;
    if (t >= rem * 64) return;
    int i = m0 + t / 64;
    int j = t % 64;
    float acc = b2[j];
    for (int h = 0; h < 64; ++h) {
        float s = b1[h];
        for (int k = 0; k < 16; ++k) s += W1[h * 16 + k] * feat[(size_t)k * N + i];
        s = s > 0.0f ? s : 0.0f;
        acc += W2[j * 64 + h] * s;
    }
    out[(size_t)i * 64 + j] = acc;
}

// ---------------------------------------------------------------------------
// Host orchestration (graph-capture safe: everything on `stream`, sizes from
// in_sizes/out_size only — N = out_size/64, E = len(edge_weight)).
// ---------------------------------------------------------------------------
extern "C" void kernel_launch(void* const* d_in, const int* in_sizes, int n_in,
                              void* d_out, int out_size, void* d_ws, size_t ws_size,
                              hipStream_t stream) {
    (void)n_in; (void)ws_size;
    const int*   edge_index = (const int*)d_in[0];
    const float* ew         = (const float*)d_in[2];
    const float* W1         = (const float*)d_in[3];
    const float* b1         = (const float*)d_in[4];
    const float* W2         = (const float*)d_in[5];
    const float* b2         = (const float*)d_in[6];
    float*       out        = (float*)d_out;

    const int E = in_sizes[2];          // edge_weight length
    const int N = out_size / 64;        // output is (N, 64)
    const int* row = edge_index;        // edge_index is (2, E) row-major
    const int* col = edge_index + E;

    float* ws      = (float*)d_ws;
    float* deg_inv = ws;                // N floats
    float* probs   = ws + N;            // 17 rows x N floats (row 0 = init)

    const int gN = (N + 255) / 256;
    const int gE = (E + 255) / 256;

    k_init_deg  <<<gN, 256, 0, stream>>>(deg_inv, N);
    k_accum_deg <<<gE, 256, 0, stream>>>(row, ew, deg_inv, E);
    k_finish_deg<<<gN, 256, 0, stream>>>(deg_inv, probs, N);

    for (int k = 1; k <= 16; ++k) {
        const float* prev = probs + (size_t)(k - 1) * N;
        float*       next = probs + (size_t)k * N;
        k_self_step<<<gN, 256, 0, stream>>>(prev, deg_inv, next, N);
        k_scatter  <<<gE, 256, 0, stream>>>(row, col, ew, deg_inv, prev, next, E);
    }

    const float* feat = probs + N;      // rows 1..16 => (16, N) features
    const int tiles = N / 16;
    if (tiles > 0)
        rwpe_mlp_wmma<<<tiles, 32, 0, stream>>>(feat, W1, b1, W2, b2, out, N);
    const int rem = N - tiles * 16;
    if (rem > 0)
        mlp_tail<<<(rem * 64 + 255) / 256, 256, 0, stream>>>(feat, W1, b1, W2, b2,
                                                             out, N, tiles * 16);
}